// QuantizerEMA_36335423324705
// MI455X (gfx1250) — compile-verified
//
#include <hip/hip_runtime.h>
#include <hip/hip_bf16.h>

#define DECAY_  0.99f
#define COMMIT_ 0.25f
#define EPS_    1e-5f

typedef float v2f __attribute__((ext_vector_type(2)));
typedef float v8f __attribute__((ext_vector_type(8)));

// ---------------------------------------------------------------------------
// Kernel 0: enorm[k] = ||embeddings[k]||^2   (K=1024, D=64)
// ---------------------------------------------------------------------------
__global__ void vq_enorm(const float* __restrict__ E, float* __restrict__ enorm) {
    int k = blockIdx.x * blockDim.x + threadIdx.x;
    if (k >= 1024) return;
    const float4* row = reinterpret_cast<const float4*>(E + (size_t)k * 64);
    float s = 0.0f;
#pragma unroll
    for (int i = 0; i < 16; ++i) {
        float4 v = row[i];
        s += v.x * v.x + v.y * v.y + v.z * v.z + v.w * v.w;
    }
    enorm[k] = s;
}

// ---------------------------------------------------------------------------
// Async copy helper: one lane moves 16B global -> LDS (ASYNCcnt-tracked DMA).
// lds_off is the LDS byte offset (low 32 bits of a generic LDS pointer).
// ---------------------------------------------------------------------------
__device__ __forceinline__ void async_b128(uint32_t lds_off, const float* gsrc) {
    asm volatile("global_load_async_to_lds_b128 %0, %1, off"
                 :: "v"(lds_off), "v"(gsrc) : "memory");
}
__device__ __forceinline__ void wait_async0() {
    asm volatile("s_wait_asynccnt 0x0" ::: "memory");
}

// Stage one 16-col x 64-deep B tile (4KB) with 128 threads: 32B per thread.
__device__ __forceinline__ void stage_tile(const float* __restrict__ E, int col0,
                                           uint32_t lds_dst_base, int tid) {
    int j  = tid >> 3;          // 0..15 : embedding column within tile
    int f0 = (tid & 7) << 3;    // 0,8,...,56 : float offset within row
    const float* src = E + (((size_t)(col0 + j)) << 6) + f0;
    uint32_t dst = lds_dst_base + (uint32_t)(((j << 6) + f0) << 2);
    async_b128(dst,       src);
    async_b128(dst + 16u, src + 4);
}

// ---------------------------------------------------------------------------
// Kernel 1: fused distance + argmin via V_WMMA_F32_16X16X4_F32.
// dist_rel[n,k] = ||e_k||^2 - 2 * z_n . e_k   (||z_n||^2 is row-constant)
// A is preloaded as (-2 * z) so the WMMA accumulates directly onto enorm.
// Block = 128 threads = 4 waves; each wave owns TWO 16-row tiles (128 rows
// per block). B tiles are DMA'd into double-buffered LDS once per block and
// shared by all 4 waves / both row tiles -> 8x less L2 traffic per row.
// ---------------------------------------------------------------------------
__global__ void vq_argmin(const float* __restrict__ z,
                          const float* __restrict__ E,
                          const float* __restrict__ enorm,
                          int* __restrict__ idx_out) {
    __shared__ float smemB[2][16][64];   // [buf][col][d]  (2 x 4KB)

    const int tid   = threadIdx.x;
    const int lane  = tid & 31;
    const int wave  = tid >> 5;
    const int m     = lane & 15;          // row (A) / col (B,C) within tile
    const int koff  = (lane >> 4) << 1;   // K sub-offset: 0 (lanes 0-15) / 2 (16-31)
    const int n_base0 = blockIdx.x * 128 + wave * 32;        // tile t adds t*16

    const uint32_t lds_base = (uint32_t)(uintptr_t)(&smemB[0][0][0]);

    // Preload A = -2*z for both 16x64 row tiles (16 chunks of 16x4 each).
    v2f a[2][16];
#pragma unroll
    for (int t = 0; t < 2; ++t) {
        const float* zrow = z + (size_t)(n_base0 + t * 16 + m) * 64;
#pragma unroll
        for (int c = 0; c < 16; ++c) {
            v2f av = *reinterpret_cast<const v2f*>(zrow + 4 * c + koff);
            a[t][c].x = -2.0f * av.x;
            a[t][c].y = -2.0f * av.y;
        }
    }

    float best[2][8];
    int   bidx[2][8];
#pragma unroll
    for (int t = 0; t < 2; ++t)
#pragma unroll
        for (int r = 0; r < 8; ++r) { best[t][r] = 3.4e38f; bidx[t][r] = 0; }

    // Prologue: DMA first B tile into buffer 0.
    stage_tile(E, 0, lds_base, tid);
    wait_async0();
    __syncthreads();

    for (int kt = 0; kt < 64; ++kt) {
        const int cur  = kt & 1;
        const int col0 = kt * 16;
        const int kcol = col0 + m;           // column this lane's C slots hold
        const float en = enorm[kcol];

        // Kick off DMA of the next tile into the other buffer.
        if (kt < 63)
            stage_tile(E, col0 + 16, lds_base + (uint32_t)((cur ^ 1) * 4096), tid);

        // Pull this lane's B values from LDS once; reuse for both row tiles.
        v2f breg[16];
#pragma unroll
        for (int c = 0; c < 16; ++c)
            breg[c] = *reinterpret_cast<const v2f*>(&smemB[cur][m][4 * c + koff]);

#pragma unroll
        for (int t = 0; t < 2; ++t) {
            v8f cacc;
#pragma unroll
            for (int r = 0; r < 8; ++r) cacc[r] = en;   // init C with ||e||^2
#pragma unroll
            for (int c = 0; c < 16; ++c) {
                cacc = __builtin_amdgcn_wmma_f32_16x16x4_f32(
                    /*neg_a=*/false, a[t][c], /*neg_b=*/false, breg[c],
                    /*c_mod=*/(short)0, cacc, /*reuse_a=*/false, /*reuse_b=*/false);
            }
            // Running per-slot min: slot (r,lane) always sees column kcol and
            // kcol grows monotonically -> strict '<' keeps the smallest k.
#pragma unroll
            for (int r = 0; r < 8; ++r) {
                float dv = cacc[r];
                if (dv < best[t][r]) { best[t][r] = dv; bidx[t][r] = kcol; }
            }
        }

        wait_async0();      // our DMA into the other buffer has landed
        __syncthreads();    // everyone done reading cur; next iter overwrites it
    }

    // Cross-lane reduce within each 16-lane half (C layout: lanes 0-15 hold
    // rows r, lanes 16-31 hold rows r+8; xor masks < 16 stay in the half).
#pragma unroll
    for (int t = 0; t < 2; ++t) {
#pragma unroll
        for (int r = 0; r < 8; ++r) {
            float v = best[t][r];
            int   i = bidx[t][r];
#pragma unroll
            for (int off = 8; off >= 1; off >>= 1) {
                float ov = __shfl_xor(v, off, 32);
                int   oi = __shfl_xor(i, off, 32);
                if (ov < v || (ov == v && oi < i)) { v = ov; i = oi; }
            }
            if (m == 0) {
                int row = r + ((lane >> 4) << 3);
                idx_out[n_base0 + t * 16 + row] = i;
            }
        }
    }
}

// ---------------------------------------------------------------------------
// Kernel 2: gather + straight-through output with LDS transpose to [B,D,H,W],
// index output, loss partials, segment sums (dw, n_i) via atomics.
// Block = 256 threads handles 32 consecutive n (one (b,h) strip) x D=64.
// ---------------------------------------------------------------------------
__global__ void vq_gather(const float* __restrict__ z,
                          const float* __restrict__ E,
                          const int* __restrict__ idx,
                          float* __restrict__ out_qv,
                          float* __restrict__ out_idx,
                          float* __restrict__ ws_dw,
                          float* __restrict__ ws_ni,
                          float* __restrict__ ws_losspart) {
    __shared__ float lds[32][65];
    __shared__ float lsum[8];
    const int tid = threadIdx.x;
    const int n0  = blockIdx.x * 32;

    float lp = 0.0f;
#pragma unroll
    for (int i = 0; i < 8; ++i) {
        int e   = i * 256 + tid;
        int row = e >> 6;
        int col = e & 63;
        int n   = n0 + row;
        int k   = idx[n];
        float zv = z[(size_t)n * 64 + col];
        float ev = E[(size_t)k * 64 + col];
        float ste = zv + (ev - zv);            // matches z + sg(q - z) bit-for-bit
        lds[row][col] = ste;
        float diff = zv - ev;
        lp += diff * diff;
        atomicAdd(&ws_dw[(size_t)k * 64 + col], zv);
        if (col == 0) {
            atomicAdd(&ws_ni[k], 1.0f);        // integer-valued float: exact
            out_idx[n] = (float)k;
        }
    }
    __syncthreads();

    // deterministic per-block loss partial
#pragma unroll
    for (int off = 16; off >= 1; off >>= 1) lp += __shfl_xor(lp, off, 32);
    if ((tid & 31) == 0) lsum[tid >> 5] = lp;
    __syncthreads();
    if (tid == 0) {
        float s = 0.0f;
#pragma unroll
        for (int wv = 0; wv < 8; ++wv) s += lsum[wv];
        ws_losspart[blockIdx.x] = s;
    }

    // transposed write: n0 = b*1024 + h*32 ; out[b, d, h, w]
    const int b = blockIdx.x >> 5;
    const int h = blockIdx.x & 31;
    float* outb = out_qv + (size_t)b * 64 * 1024 + (size_t)h * 32;
#pragma unroll
    for (int i = 0; i < 8; ++i) {
        int e = i * 256 + tid;
        int d = e >> 5;
        int w = e & 31;
        outb[(size_t)d * 1024 + w] = lds[w][d];
    }
}

// ---------------------------------------------------------------------------
// Kernel 3a: cluster EMA + Laplace smoothing + loss finalize. 1 block x 1024.
// ---------------------------------------------------------------------------
__global__ void vq_cluster(const float* __restrict__ cs,
                           const float* __restrict__ ni,
                           const float* __restrict__ losspart,
                           float* __restrict__ out_cluster,
                           float* __restrict__ out_loss,
                           float* __restrict__ ws_cluster) {
    __shared__ float redA[32], redB[32];
    __shared__ float nTot, lossTot;
    const int k = threadIdx.x;

    float ncp = cs[k] * DECAY_ + ni[k] * (1.0f - DECAY_);
    float a = ncp;
    float b = losspart[k];
#pragma unroll
    for (int off = 16; off >= 1; off >>= 1) {
        a += __shfl_xor(a, off, 32);
        b += __shfl_xor(b, off, 32);
    }
    if ((k & 31) == 0) { redA[k >> 5] = a; redB[k >> 5] = b; }
    __syncthreads();
    if (k < 32) {
        float a2 = redA[k], b2 = redB[k];
#pragma unroll
        for (int off = 16; off >= 1; off >>= 1) {
            a2 += __shfl_xor(a2, off, 32);
            b2 += __shfl_xor(b2, off, 32);
        }
        if (k == 0) { nTot = a2; lossTot = b2; }
    }
    __syncthreads();

    float n = nTot;
    float ncl = (ncp + EPS_) / (n + 1024.0f * EPS_) * n;
    out_cluster[k] = ncl;
    ws_cluster[k]  = ncl;
    if (k == 0) out_loss[0] = COMMIT_ * lossTot / (float)(32768 * 64);
}

// ---------------------------------------------------------------------------
// Kernel 3b: new_ema / new_embeddings, flat over K*D = 65536.
// ---------------------------------------------------------------------------
__global__ void vq_embed(const float* __restrict__ ema,
                         const float* __restrict__ ws_dw,
                         const float* __restrict__ ws_cluster,
                         float* __restrict__ out_ema,
                         float* __restrict__ out_emb) {
    int i = blockIdx.x * blockDim.x + threadIdx.x;
    if (i >= 65536) return;
    int k = i >> 6;
    float e = ema[i] * DECAY_ + ws_dw[i] * (1.0f - DECAY_);
    out_ema[i] = e;
    out_emb[i] = e / ws_cluster[k];
}

// ---------------------------------------------------------------------------
extern "C" void kernel_launch(void* const* d_in, const int* in_sizes, int n_in,
                              void* d_out, int out_size, void* d_ws, size_t ws_size,
                              hipStream_t stream) {
    const float* z   = (const float*)d_in[0];  // [32,32,32,64]
    const float* E   = (const float*)d_in[1];  // [1024,64]
    const float* cs  = (const float*)d_in[2];  // [1024]
    const float* ema = (const float*)d_in[3];  // [1024,64]
    float* out = (float*)d_out;
    char*  ws  = (char*)d_ws;

    // workspace layout (bytes)
    int*   ws_idx      = (int*)  (ws + 0);        // 32768 ints
    float* ws_enorm    = (float*)(ws + 131072);   // 1024
    float* ws_ni       = (float*)(ws + 135168);   // 1024   (zeroed)
    float* ws_dw       = (float*)(ws + 139264);   // 65536  (zeroed, contiguous w/ ni)
    float* ws_losspart = (float*)(ws + 401408);   // 1024
    float* ws_cluster  = (float*)(ws + 405504);   // 1024

    // output layout (floats)
    float* out_qv      = out + 0;          // 2097152  [B,D,H,W]
    float* out_idx     = out + 2097152;    // 32768    [B,1,H,W]
    float* out_loss    = out + 2129920;    // 1
    float* out_emb     = out + 2129921;    // 65536
    float* out_cluster = out + 2195457;    // 1024
    float* out_ema     = out + 2196481;    // 65536

    hipMemsetAsync(ws_ni, 0, (1024 + 65536) * sizeof(float), stream);

    vq_enorm  <<<4,    256, 0, stream>>>(E, ws_enorm);
    vq_argmin <<<256,  128, 0, stream>>>(z, E, ws_enorm, ws_idx);
    vq_gather <<<1024, 256, 0, stream>>>(z, E, ws_idx, out_qv, out_idx,
                                         ws_dw, ws_ni, ws_losspart);
    vq_cluster<<<1,   1024, 0, stream>>>(cs, ws_ni, ws_losspart,
                                         out_cluster, out_loss, ws_cluster);
    vq_embed  <<<256,  256, 0, stream>>>(ema, ws_dw, ws_cluster, out_ema, out_emb);
}